// Multi_Layer_Renderer_40235253629025
// MI455X (gfx1250) — compile-verified
//
#include <hip/hip_runtime.h>
#include <cstdint>
#include <cstddef>

// ---------------------------------------------------------------------------
// Problem constants (B=2, 3 layers, 512x512, 21x21 disk footprint, R=10)
// ---------------------------------------------------------------------------
#define BATCH   2
#define NLAYER  3
#define H_IMG   512
#define W_IMG   512
#define PH      (H_IMG + 20)       // 532 padded rows (radius-10 halo)
#define PW      536                // padded row width, 16B-aligned rows (532 used)
#define PLANE   (PH * PW)          // 285152 floats per channel plane
#define EPSF    1e-8f

#define TILE    32
#define HALO    (TILE + 20)        // 52
#define SM_PLANE (HALO * HALO)     // 2704
#define TILE_FLOATS (5 * SM_PLANE) // 13520 floats per layer tile

typedef unsigned int v4u __attribute__((ext_vector_type(4)));
typedef int          v4i __attribute__((ext_vector_type(4)));
typedef int          v8i __attribute__((ext_vector_type(8)));

// ---------------------------------------------------------------------------
// Compile-time tables (constexpr sqrt via Newton in double)
// ---------------------------------------------------------------------------
constexpr float csqrtf_ce(double x) {
  if (x <= 0.0) return 0.0f;
  double r = x;
  for (int i = 0; i < 64; ++i) r = 0.5 * (r + x / r);
  return (float)r;
}

// area(r) = sum over disk offsets of clip(r+0.5-d,0,1): collapse 441 offsets
// to the 44 distinct distances with multiplicities.
struct AreaTab { int n; float d[48]; float c[48]; };
constexpr AreaTab make_area_tab() {
  AreaTab t{};
  int cnt[101] = {};
  for (int dy = -10; dy <= 10; ++dy)
    for (int dx = -10; dx <= 10; ++dx) {
      int q = dy * dy + dx * dx;
      if (q <= 100) cnt[q]++;
    }
  t.n = 0;
  for (int q = 0; q <= 100; ++q)
    if (cnt[q]) { t.d[t.n] = csqrtf_ce((double)q); t.c[t.n] = (float)cnt[q]; t.n++; }
  return t;
}
constexpr AreaTab kAreaTab = make_area_tab();
constexpr int NAREA = kAreaTab.n;   // 44

// Full 21x21 distance table + per-row disk bounds (mask d<=10 is a contiguous
// dx range per dy).
struct DistTab { float d[441]; int xmin[21]; int xmax[21]; };
constexpr DistTab make_dist_tab() {
  DistTab t{};
  for (int dy = 0; dy < 21; ++dy) {
    int rdy = dy - 10;
    int lim = 100 - rdy * rdy;
    int m = 0;
    while ((m + 1) * (m + 1) <= lim) ++m;
    t.xmin[dy] = 10 - m;
    t.xmax[dy] = 10 + m;
    for (int dx = 0; dx < 21; ++dx) {
      int rdx = dx - 10;
      t.d[dy * 21 + dx] = csqrtf_ce((double)(rdy * rdy + rdx * rdx));
    }
  }
  return t;
}
__constant__ AreaTab g_area = make_area_tab();
__constant__ DistTab g_dist = make_dist_tab();

// ---------------------------------------------------------------------------
// CDNA5 data-movement paths (best available wins):
//   1) TDM tensor_load_to_lds: one 3D-tile DMA per layer, TENSORcnt tracked
//   2) per-lane global_load_async_to_lds_b128 (proven to compile, round 2)
//   3) plain b128 load + ds_store
// AS pointers built via integer round-trips: AS1 keeps the flat 64-bit value;
// AS3 is 32-bit and flat-shared low bits ARE the LDS offset (ISA 10.2).
// ---------------------------------------------------------------------------
#if __has_builtin(__builtin_amdgcn_tensor_load_to_lds) && \
    __has_builtin(__builtin_amdgcn_s_wait_tensorcnt)
#define USE_TDM 1
#endif
#if __has_builtin(__builtin_amdgcn_global_load_async_to_lds_b128) && \
    __has_builtin(__builtin_amdgcn_s_wait_asynccnt)
#define USE_ASYNC_LDS 1
typedef __attribute__((address_space(1))) v4i* Gv4iPtr;
typedef __attribute__((address_space(3))) v4i* Lv4iPtr;
#define ASYNC_CPY_B128(gp_, lp_)                                           \
  __builtin_amdgcn_global_load_async_to_lds_b128(                          \
      (Gv4iPtr)(unsigned long long)(uintptr_t)(gp_),                       \
      (Lv4iPtr)(unsigned int)(uintptr_t)(lp_), 0, 0)
#endif

#if defined(USE_TDM)
// One TDM descriptor: 3D tile (x=52, y=52, z=5 channels), element 4B,
// strides: row = PW floats, channel = PLANE floats. gp0 = tile start.
// This toolchain's builtin is the 6-arg form:
//   (v4u group0, v8i group1, v4i group2, v4i group3, v8i unused, i32 cpol)
__device__ __forceinline__ void tdm_issue_tile(const float* gp0, float* ldst) {
  unsigned long long ga = (unsigned long long)(uintptr_t)gp0;
  unsigned int lds = (unsigned int)(uintptr_t)ldst;
  v4u g0;
  g0.x = 1u;                                         // count=1 (valid), user mode
  g0.y = lds;                                        // lds_addr (bytes)
  g0.z = (unsigned int)(ga & 0xFFFFFFFFu);           // global_addr[31:0]
  g0.w = (unsigned int)((ga >> 32) & 0x01FFFFFFu)    // global_addr[56:32]
         | (2u << 30);                               // type=2 ("image")
  v8i g1;
  g1[0] = 0x20000;                                   // data_size=2 (4 bytes)
  g1[1] = (int)((536u & 0xFFFFu) << 16);             // tensor_dim0 lo16 @bits63:48
  g1[2] = (int)((536u >> 16) | ((532u & 0xFFFFu) << 16)); // dim0 hi16 | dim1 lo16
  g1[3] = (int)((532u >> 16) | (52u << 16));         // dim1 hi16 | tile_dim0=52
  g1[4] = (int)(52u | (5u << 16));                   // tile_dim1=52 | tile_dim2=5
  g1[5] = 536;                                       // tensor_dim0_stride lo32
  g1[6] = (int)((((unsigned)PLANE) & 0xFFFFu) << 16);// dim0str hi16=0 | dim1str lo16
  g1[7] = (int)(((unsigned)PLANE) >> 16);            // dim1_stride bits 47:16
  v4i g2;
  g2[0] = 5;                                         // tensor_dim2
  g2[1] = 0;                                         // tensor_dim3 (unused)
  g2[2] = PLANE;                                     // tensor_dim2_stride lo32
  g2[3] = 0;                                         // dim2str hi16=0 | tile_dim3=0
  v4i g3; g3[0] = 0; g3[1] = 0; g3[2] = 0; g3[3] = 0;
  v8i g4; g4[0] = 0; g4[1] = 0; g4[2] = 0; g4[3] = 0;
          g4[4] = 0; g4[5] = 0; g4[6] = 0; g4[7] = 0; // VADDR4 slot: unused
  __builtin_amdgcn_tensor_load_to_lds(g0, g1, g2, g3, g4, 0);
}
#endif

// Issue the (asynchronous) fill of one 5x52x52 layer tile into LDS.
__device__ __forceinline__ void tile_issue(const float* plane0, int y0, int x0,
                                           float* dst, int tid) {
  const float* gp0 = plane0 + (size_t)y0 * PW + x0;
#if defined(USE_TDM)
  if (tid < 32) tdm_issue_tile(gp0, dst);
#else
  for (int q = tid; q < 5 * 52 * 13; q += 256) {     // 16B-aligned quads
    int c    = q / (52 * 13);
    int rm   = q - c * (52 * 13);
    int row  = rm / 13;
    int quad = rm - row * 13;
    const float* gp = gp0 + (size_t)c * PLANE + (size_t)row * PW + quad * 4;
    float* lp = dst + c * SM_PLANE + row * HALO + quad * 4;
#if defined(USE_ASYNC_LDS)
    ASYNC_CPY_B128(gp, lp);
#else
    const float4 v = *(const float4*)gp;
    *(float4*)lp = v;
#endif
  }
#endif
}

__device__ __forceinline__ void tile_wait(int tid) {
#if defined(USE_TDM)
  if (tid < 32) __builtin_amdgcn_s_wait_tensorcnt(0);
#elif defined(USE_ASYNC_LDS)
  __builtin_amdgcn_s_wait_asynccnt(0);
#endif
}

// ---------------------------------------------------------------------------
// Kernel 1: per-pixel preprocess into zero-padded SoA scratch
//   ws layout: [B][L][5][PH][PW] float; channels 0..3 = premultiplied s,
//   channel 4 = r + 0.5 (padding regions written as zero)
// ---------------------------------------------------------------------------
__global__ __launch_bounds__(256)
void bokeh_prep_kernel(const float* __restrict__ rgbad,
                       const float* __restrict__ lens_eff,
                       const float* __restrict__ focal,
                       float* __restrict__ ws) {
  int t = blockIdx.x * 256 + threadIdx.x;
  const int total = BATCH * NLAYER * PH * PW;
  if (t >= total) return;
  int px = t % PW; int r1 = t / PW;
  int py = r1 % PH; int r2 = r1 / PH;
  int l  = r2 % NLAYER;
  int b  = r2 / NLAYER;

  float* wp = ws + (size_t)((b * NLAYER + l) * 5) * PLANE + (size_t)py * PW + px;

  int ix = px - 10, iy = py - 10;
  if ((unsigned)ix >= (unsigned)W_IMG || (unsigned)iy >= (unsigned)H_IMG) {
    wp[0 * PLANE] = 0.f; wp[1 * PLANE] = 0.f; wp[2 * PLANE] = 0.f;
    wp[3 * PLANE] = 0.f; wp[4 * PLANE] = 0.f;
    return;
  }

  const size_t cs = (size_t)H_IMG * W_IMG;
  size_t gi = ((size_t)((b * NLAYER + l) * 5) * H_IMG + iy) * W_IMG + ix;
  float r  = rgbad[gi]          - focal[gi];
  float g  = rgbad[gi + cs]     - focal[gi + cs];
  float bl = rgbad[gi + 2 * cs] - focal[gi + 2 * cs];
  float a  = rgbad[gi + 3 * cs] - focal[gi + 3 * cs];
  float dp = rgbad[gi + 4 * cs] - focal[gi + 4 * cs];

  float lens = lens_eff[b];                       // [B,1]
  float rr = fminf(fmaxf(fabsf(dp) * lens, 0.f), 10.f);

  float area = 0.f;
#pragma unroll
  for (int k = 0; k < NAREA; ++k)
    area += g_area.c[k] * fminf(fmaxf(rr + 0.5f - g_area.d[k], 0.f), 1.f);

  float inv = 1.f / (area + EPSF);
  wp[0 * PLANE] = r  * a * inv;
  wp[1 * PLANE] = g  * a * inv;
  wp[2 * PLANE] = bl * a * inv;
  wp[3 * PLANE] = a * inv;
  wp[4 * PLANE] = rr + 0.5f;
}

// ---------------------------------------------------------------------------
// Kernel 2: double-buffered LDS-tiled gather + front-to-back composite
//   grid = (16,16,B); block = 256 (8 waves of 32); each thread -> 4 pixels
//   DMA of layer l+1 overlaps compute of layer l.
// ---------------------------------------------------------------------------
__global__ __launch_bounds__(256)
void bokeh_gather_kernel(const float* __restrict__ ws,
                         float* __restrict__ out) {
  extern __shared__ float smem[];                 // 2 * TILE_FLOATS floats

  const int tid = threadIdx.x;
  const int lx  = tid & 31;                       // output col in tile
  const int tyi = tid >> 5;                       // wave id 0..7 -> base row
  const int x0  = blockIdx.x * TILE;
  const int y0  = blockIdx.y * TILE;
  const int b   = blockIdx.z;

  float* buf0 = smem;
  float* buf1 = smem + TILE_FLOATS;

  float blurR[4], blurG[4], blurB[4], trans[4];
#pragma unroll
  for (int k = 0; k < 4; ++k) { blurR[k] = blurG[k] = blurB[k] = 0.f; trans[k] = 1.f; }

  const float* base = ws + (size_t)(b * NLAYER) * 5 * PLANE;

  // Prologue: fetch layer 0 into buf0.
  tile_issue(base, y0, x0, buf0, tid);
  tile_wait(tid);
  __syncthreads();

  for (int l = 0; l < NLAYER; ++l) {
    float* cur = (l & 1) ? buf1 : buf0;
    float* nxt = (l & 1) ? buf0 : buf1;

    // Kick off DMA of the next layer while we compute this one.
    if (l + 1 < NLAYER)
      tile_issue(base + (size_t)(l + 1) * 5 * PLANE, y0, x0, nxt, tid);

    // Warm L2 for the layer after next (emits global_prefetch_b8).
    if (l + 2 < NLAYER) {
      const float* np = base + (size_t)(l + 2) * 5 * PLANE +
                        (size_t)(y0 + (tid >> 3)) * PW + x0 + (tid & 7) * 8;
      __builtin_prefetch(np, 0, 1);
    }

#pragma unroll
    for (int k = 0; k < 4; ++k) {
      const int ly = tyi + 8 * k;                 // output row 0..31 in tile
      float a0 = 0.f, a1 = 0.f, a2 = 0.f, a3 = 0.f;
      for (int dy = 0; dy < 21; ++dy) {
        const int   rb = (ly + dy) * HALO + lx;
        const int   xm = g_dist.xmin[dy];
        const int   xM = g_dist.xmax[dy];
        const float* dd = &g_dist.d[dy * 21];     // uniform -> scalar loads
        for (int dx = xm; dx <= xM; ++dx) {
          const int si = rb + dx;
          float w = fminf(fmaxf(cur[4 * SM_PLANE + si] - dd[dx], 0.f), 1.f);
          a0 = fmaf(cur[0 * SM_PLANE + si], w, a0);
          a1 = fmaf(cur[1 * SM_PLANE + si], w, a1);
          a2 = fmaf(cur[2 * SM_PLANE + si], w, a2);
          a3 = fmaf(cur[3 * SM_PLANE + si], w, a3);
        }
      }
      // composite: blur += rgb/(w+eps) * (trans*occu); trans *= (1-occu)
      float o   = fminf(fmaxf(a3, 0.f), 1.f);
      float inv = 1.f / (a3 + EPSF);
      float f   = trans[k] * o * inv;
      blurR[k] = fmaf(a0, f, blurR[k]);
      blurG[k] = fmaf(a1, f, blurG[k]);
      blurB[k] = fmaf(a2, f, blurB[k]);
      trans[k] *= (1.f - o);
    }

    __syncthreads();                              // all reads of cur done
    if (l + 1 < NLAYER) {
      tile_wait(tid);                             // next buffer complete
      __syncthreads();                            // publish to all waves
    }
  }

  // out: [B][3][H][W]
  const size_t cs = (size_t)H_IMG * W_IMG;
#pragma unroll
  for (int k = 0; k < 4; ++k) {
    int y = y0 + tyi + 8 * k;
    int x = x0 + lx;
    size_t o0 = ((size_t)(b * 3) * H_IMG + y) * W_IMG + x;
    out[o0]          = blurR[k];
    out[o0 + cs]     = blurG[k];
    out[o0 + 2 * cs] = blurB[k];
  }
}

// ---------------------------------------------------------------------------
// Launch: inputs = {rgbad_layers[B,15,H,W], lens_effect[B,1], focal[B,15,H,W]}
// ws usage: B*L*5*PH*PW floats = 34,218,240 bytes
// ---------------------------------------------------------------------------
extern "C" void kernel_launch(void* const* d_in, const int* in_sizes, int n_in,
                              void* d_out, int out_size, void* d_ws, size_t ws_size,
                              hipStream_t stream) {
  (void)in_sizes; (void)n_in; (void)out_size; (void)ws_size;
  const float* rgbad = (const float*)d_in[0];
  const float* lens  = (const float*)d_in[1];
  const float* focal = (const float*)d_in[2];
  float* ws  = (float*)d_ws;
  float* out = (float*)d_out;

  const int totalPrep = BATCH * NLAYER * PH * PW;            // 1,710,912
  bokeh_prep_kernel<<<(totalPrep + 255) / 256, 256, 0, stream>>>(rgbad, lens, focal, ws);

  dim3 grid(W_IMG / TILE, H_IMG / TILE, BATCH);              // (16,16,2)
  const size_t shmem = 2 * TILE_FLOATS * sizeof(float);      // 108,160 B
  bokeh_gather_kernel<<<grid, 256, shmem, stream>>>(ws, out);
}